// SequenceGeneratorFuck_70944269795502
// MI455X (gfx1250) — compile-verified
//
#include <hip/hip_runtime.h>
#include <hip/hip_bf16.h>
#include <math.h>
#include <stdint.h>

// ---------------------------------------------------------------------------
// Beam-search decode for MI455X (gfx1250).
//
// Roofline: 262 MB of logits / 23.3 TB/s => ~11 us floor; compute trivial
// (1 exp + compare per element). No matmul content -> WMMA inapplicable; the
// CDNA5 features that pay are the Tensor Data Mover (async global->LDS,
// double buffered, TENSORcnt-pipelined) and wave32 shuffle reductions.
//
// Pass 1 (bandwidth bound, 2048 blocks): per-row online logsumexp + top-8 of
//   raw logits (same ordering as log_softmax), streamed through LDS by the
//   TDM in 25.6 KB chunks, consumed with ds_load_b128.
// Pass 2 (negligible): sequential 16-step beam recurrence on 4x8 candidates.
// ---------------------------------------------------------------------------

#define T_STEPS 16
#define BBDIM   128
#define VOCAB   32000
#define BEAM    4
#define BATCH   (BBDIM / BEAM)      // 32
#define CHUNK   6400                // floats per TDM chunk (25.6 KB)
#define NCHUNK  (VOCAB / CHUNK)     // 5
#define NT1     256                 // pass-1 block size (8 waves of 32)
#define NWAVE   (NT1 / 32)

typedef unsigned int  v4u __attribute__((ext_vector_type(4)));
typedef int           v8i __attribute__((ext_vector_type(8)));
typedef int           v4i __attribute__((ext_vector_type(4)));

#if defined(__has_builtin)
#  if __has_builtin(__builtin_amdgcn_tensor_load_to_lds)
#    define HAVE_TDM 1
#  endif
#endif
#ifndef HAVE_TDM
#  define HAVE_TDM 0
#endif

// Monotone float -> uint mapping (max-reduce friendly, order preserving).
__device__ __forceinline__ unsigned ordf(float f) {
  unsigned u = __float_as_uint(f);
  return u ^ ((unsigned)((int)u >> 31) | 0x80000000u);
}
__device__ __forceinline__ float unordf(unsigned u) {
  u ^= ((u >> 31) ? 0x80000000u : 0xFFFFFFFFu);
  return __uint_as_float(u);
}
// key: high32 = ordered score, low32 = ~flat_index (ties -> lower index wins)
__device__ __forceinline__ unsigned long long packKey(float f, int idx) {
  return ((unsigned long long)ordf(f) << 32) | (unsigned)(~(unsigned)idx);
}

__device__ __forceinline__ unsigned long long shflxor64(unsigned long long v,
                                                        int off) {
  unsigned lo = (unsigned)(v & 0xFFFFFFFFull);
  unsigned hi = (unsigned)(v >> 32);
  lo = __shfl_xor(lo, off, 32);
  hi = __shfl_xor(hi, off, 32);
  return ((unsigned long long)hi << 32) | lo;
}

// Issue one 1-D TDM copy: nElems f32 from global addr -> LDS byte offset.
// D# per CDNA5 ISA.
__device__ __forceinline__ void tdm_load_1d(unsigned long long gaddr,
                                            unsigned lds_addr, unsigned nElems) {
#if HAVE_TDM
  v4u g0;
  g0.x = 1u;                                      // count=1 (valid user D#)
  g0.y = lds_addr;                                // lds_addr[31:0]
  g0.z = (unsigned)(gaddr & 0xFFFFFFFFull);       // global_addr[31:0]
  g0.w = (unsigned)((gaddr >> 32) & 0x1FFFFFFull) // global_addr[56:32]
         | (2u << 30);                            // type=2 ("image")
  v8i g1;
  g1[0] = (int)(2u << 16);                        // wg_mask=0, data_size=2 (4B)
  g1[1] = (int)((nElems & 0xFFFFu) << 16);        // tensor_dim0[15:0]
  g1[2] = (int)(1u << 16);                        // tensor_dim0 hi=0, tensor_dim1=1
  g1[3] = (int)((nElems & 0xFFFFu) << 16);        // tensor_dim1 hi=0, tile_dim0
  g1[4] = 1;                                      // tile_dim1=1, tile_dim2=0
  g1[5] = (int)nElems;                            // tensor_dim0_stride[31:0]
  g1[6] = 0;                                      // stride hi / dim1_stride lo
  g1[7] = 0;
  v4i g2 = {0, 0, 0, 0};
  v4i g3 = {0, 0, 0, 0};
#  if __has_include(<hip/amd_detail/amd_gfx1250_TDM.h>)
  v8i g4 = {0, 0, 0, 0, 0, 0, 0, 0};              // therock / clang-23: 6-arg
  __builtin_amdgcn_tensor_load_to_lds(g0, g1, g2, g3, g4, 0);
#  else
  __builtin_amdgcn_tensor_load_to_lds(g0, g1, g2, g3, 0);  // ROCm 7.2: 5-arg
#  endif
#else
  (void)gaddr; (void)lds_addr; (void)nElems;
#endif
}

// ---------------------------------------------------------------------------
// Pass 1: one block per row (T*BB = 2048 rows).
// ---------------------------------------------------------------------------
__global__ void __launch_bounds__(NT1)
beam_pass1_rowstats(const float* __restrict__ logits,
                    const int* __restrict__ padP, const int* __restrict__ unkP,
                    float* __restrict__ wsVal, int* __restrict__ wsIdx) {
  const int row  = blockIdx.x;
  const int tid  = threadIdx.x;
  const int lane = tid & 31;
  const int wave = tid >> 5;
  const int padI = padP[0];
  const int unkI = unkP[0];

  extern __shared__ __align__(16) float dynbuf[];   // 2 * CHUNK floats
  __shared__ float              wm[NWAVE];
  __shared__ float              wsum[NWAVE];
  __shared__ unsigned long long wkey[NWAVE];
  __shared__ unsigned long long top8[8];
  __shared__ float              lseSh;
  __shared__ unsigned long long winnerSh;

  const float* rowp = logits + (size_t)row * VOCAB;
  const unsigned ldsBase = (__builtin_amdgcn_groupstaticsize() + 15u) & ~15u;

  float m = -INFINITY, s = 0.0f;
  unsigned long long tl[8];
#pragma unroll
  for (int j = 0; j < 8; ++j) tl[j] = 0ull;

  const bool w0 = (wave == 0);   // wave 0 drives the TDM
#if HAVE_TDM
  if (w0) tdm_load_1d((unsigned long long)(uintptr_t)rowp, ldsBase, CHUNK);
#endif

  for (int c = 0; c < NCHUNK; ++c) {
    float* buf = dynbuf + (size_t)(c & 1) * CHUNK;
#if HAVE_TDM
    if (w0) {
      if (c + 1 < NCHUNK) {     // prefetch next chunk, wait only for current
        tdm_load_1d((unsigned long long)(uintptr_t)(rowp + (size_t)(c + 1) * CHUNK),
                    ldsBase + (unsigned)(((c + 1) & 1) * CHUNK * 4), CHUNK);
        __builtin_amdgcn_s_wait_tensorcnt(1);
      } else {
        __builtin_amdgcn_s_wait_tensorcnt(0);
      }
    }
    __syncthreads();
#else
    __syncthreads();
    {
      const float4* src = (const float4*)(rowp + (size_t)c * CHUNK);
      float4* dst = (float4*)buf;
      for (int i4 = tid; i4 < CHUNK / 4; i4 += NT1) dst[i4] = src[i4];
    }
    __syncthreads();
#endif
    // Consume the chunk with 128-bit LDS loads (ds_load_b128).
    const float4* bv = (const float4*)buf;
    for (int i4 = tid; i4 < CHUNK / 4; i4 += NT1) {
      const float4 x4 = bv[i4];
      const int vb = c * CHUNK + (i4 << 2);
      const float xs[4] = {x4.x, x4.y, x4.z, x4.w};
#pragma unroll
      for (int j = 0; j < 4; ++j) {
        const float x = xs[j];
        const int v = vb + j;
        // online logsumexp
        if (x > m) { s = s * __expf(m - x) + 1.0f; m = x; }
        else       { s += __expf(x - m); }
        if (v == padI || v == unkI) continue;   // masked to -inf in reference
        const unsigned long long key = packKey(x, v);
        if (key > tl[7]) {                      // insert into sorted top-8
          tl[7] = key;
#pragma unroll
          for (int jj = 7; jj > 0; --jj)
            if (tl[jj] > tl[jj - 1]) {
              unsigned long long t2 = tl[jj - 1];
              tl[jj - 1] = tl[jj];
              tl[jj] = t2;
            }
        }
      }
    }
    __syncthreads();   // readers done before this buffer is overwritten
  }

  // --- logsumexp combine: wave32 butterfly, then tiny cross-wave merge ---
#pragma unroll
  for (int off = 16; off > 0; off >>= 1) {
    const float m2 = __shfl_xor(m, off, 32);
    const float s2 = __shfl_xor(s, off, 32);
    const float M = fmaxf(m, m2);
    s = s * __expf(m - M) + s2 * __expf(m2 - M);
    m = M;
  }
  if (lane == 0) { wm[wave] = m; wsum[wave] = s; }
  __syncthreads();
  if (tid == 0) {
    float M = wm[0], S = wsum[0];
    for (int w2 = 1; w2 < NWAVE; ++w2) {
      const float Mn = fmaxf(M, wm[w2]);
      S = S * __expf(M - Mn) + wsum[w2] * __expf(wm[w2] - Mn);
      M = Mn;
    }
    lseSh = M + __logf(S);
  }
  __syncthreads();
  const float lse = lseSh;

  // --- merge per-thread top-8 lists into block top-8 (8 rounds) ---
  for (int r = 0; r < 8; ++r) {
    unsigned long long v = tl[0];
#pragma unroll
    for (int off = 16; off > 0; off >>= 1) {
      const unsigned long long o = shflxor64(v, off);
      if (o > v) v = o;
    }
    if (lane == 0) wkey[wave] = v;
    __syncthreads();
    if (tid == 0) {
      unsigned long long w = wkey[0];
      for (int w2 = 1; w2 < NWAVE; ++w2)
        if (wkey[w2] > w) w = wkey[w2];
      winnerSh = w;
      top8[r] = w;
    }
    __syncthreads();
    const unsigned long long w = winnerSh;
    if (tl[0] == w && w != 0ull) {              // unique key -> one popper
#pragma unroll
      for (int j = 0; j < 7; ++j) tl[j] = tl[j + 1];
      tl[7] = 0ull;
    }
    __syncthreads();
  }

  if (tid < 8) {
    const unsigned long long k = top8[tid];
    const float x = unordf((unsigned)(k >> 32));
    const int v = (int)(~(unsigned)(k & 0xFFFFFFFFull));
    wsVal[(size_t)row * 8 + tid] = x - lse;     // log-softmax value
    wsIdx[(size_t)row * 8 + tid] = v;
  }
}

// ---------------------------------------------------------------------------
// Pass 2: one thread per batch. 16 sequential steps on 4x8 = 32 candidates.
// ---------------------------------------------------------------------------
__global__ void __launch_bounds__(BATCH)
beam_pass2_decode(const float* __restrict__ wsVal, const int* __restrict__ wsIdx,
                  const int* __restrict__ eosP, float* __restrict__ out) {
  const int batch = threadIdx.x;
  if (batch >= BATCH) return;
  const int eosI = eosP[0];

  float sc[BEAM] = {0.f, 0.f, 0.f, 0.f};
  int tok[BEAM][T_STEPS + 1];
  for (int b = 0; b < BEAM; ++b)
    for (int j = 0; j <= T_STEPS; ++j) tok[b][j] = eosI;  // begin token = eos

  for (int t = 0; t < T_STEPS; ++t) {
    unsigned long long cand[BEAM * 8];
    for (int b = 0; b < BEAM; ++b) {
      const int rowi = t * BBDIM + batch * BEAM + b;
      for (int k = 0; k < 8; ++k) {
        const float v = wsVal[(size_t)rowi * 8 + k] + sc[b];
        const int flat = b * VOCAB + wsIdx[(size_t)rowi * 8 + k];
        cand[b * 8 + k] = packKey(v, flat);
      }
    }
    // exact top-8 of the 32 candidates (global top-8 subset of per-beam top-8s)
    float msc[8]; int bi[8], wi[8];
    bool used[BEAM * 8];
    for (int i = 0; i < BEAM * 8; ++i) used[i] = false;
    for (int r = 0; r < 8; ++r) {
      int p = -1; unsigned long long best = 0ull;
      for (int i = 0; i < BEAM * 8; ++i)
        if (!used[i] && cand[i] > best) { best = cand[i]; p = i; }
      if (p < 0) { for (int i = 0; i < BEAM * 8; ++i) if (!used[i]) { p = i; break; } }
      used[p] = true;
      const float v = unordf((unsigned)(cand[p] >> 32));
      const int flat = (int)(~(unsigned)(cand[p] & 0xFFFFFFFFull));
      bi[r] = flat / VOCAB;
      wi[r] = flat - bi[r] * VOCAB;
      const bool eos = (wi[r] == eosI) && (v > -INFINITY);
      msc[r] = eos ? -INFINITY : v;             // finalize eos candidates
    }
    // top-4 survivors of the masked list (ties -> earlier position)
    bool used2[8]; for (int r = 0; r < 8; ++r) used2[r] = false;
    float nsc[BEAM]; int ntok[BEAM][T_STEPS + 1];
    for (int k = 0; k < BEAM; ++k) {
      int p = -1; float best = 0.f;
      for (int r = 0; r < 8; ++r)
        if (!used2[r] && (p == -1 || msc[r] > best)) { p = r; best = msc[r]; }
      used2[p] = true;
      nsc[k] = msc[p];
      const int sb = bi[p], sw = wi[p];
      for (int j = 0; j <= T_STEPS; ++j) ntok[k][j] = tok[sb][j];
      ntok[k][t + 1] = sw;
    }
    for (int k = 0; k < BEAM; ++k) {
      sc[k] = nsc[k];
      for (int j = 0; j <= T_STEPS; ++j) tok[k][j] = ntok[k][j];
    }
  }

  // d_out: [batch,beam] final scores, then [batch,beam,T+1] tokens (as float)
  for (int k = 0; k < BEAM; ++k) out[batch * BEAM + k] = sc[k];
  float* toksOut = out + BATCH * BEAM;
  for (int k = 0; k < BEAM; ++k)
    for (int j = 0; j <= T_STEPS; ++j)
      toksOut[(size_t)(batch * BEAM + k) * (T_STEPS + 1) + j] = (float)tok[k][j];
}

extern "C" void kernel_launch(void* const* d_in, const int* in_sizes, int n_in,
                              void* d_out, int out_size, void* d_ws, size_t ws_size,
                              hipStream_t stream) {
  (void)in_sizes; (void)n_in; (void)out_size; (void)ws_size;
  const float* logits = (const float*)d_in[0];
  const int*   padP   = (const int*)d_in[1];
  const int*   unkP   = (const int*)d_in[2];
  const int*   eosP   = (const int*)d_in[3];
  // d_in[4] = beam_size (structurally fixed at 4 here)
  float* out = (float*)d_out;

  // Workspace: 2048 rows * 8 (lp f32) then 2048 rows * 8 (idx i32) = 128 KB.
  float* wsVal = (float*)d_ws;
  int*   wsIdx = (int*)((char*)d_ws + (size_t)T_STEPS * BBDIM * 8 * sizeof(float));

  const size_t dynBytes = (size_t)2 * CHUNK * sizeof(float);  // 51.2 KB LDS
  beam_pass1_rowstats<<<dim3(T_STEPS * BBDIM), dim3(NT1), dynBytes, stream>>>(
      logits, padP, unkP, wsVal, wsIdx);
  beam_pass2_decode<<<dim3(1), dim3(BATCH), 0, stream>>>(wsVal, wsIdx, eosP, out);
}